// IouLoss_51221779972846
// MI455X (gfx1250) — compile-verified
//
#include <hip/hip_runtime.h>
#include <hip/hip_bf16.h>
#include <math.h>

#define EPS 1e-8f
#define N_GT 512
#define TPB 256

typedef __attribute__((ext_vector_type(2))) float v2f;
typedef __attribute__((ext_vector_type(8))) float v8f;

__global__ void IouLoss_zero_kernel(float* out) { out[0] = 0.0f; }

__global__ __launch_bounds__(TPB) void IouLoss_main_kernel(
    const float* __restrict__ iou_pred,
    const float* __restrict__ box_pred,
    const float* __restrict__ box_gt,
    float* __restrict__ out,
    int n_pred)
{
    __shared__ float s_raw[N_GT * 7];   // raw gt boxes (async-staged)
    __shared__ float s_gcx[N_GT * 4];   // gt corner x
    __shared__ float s_gcy[N_GT * 4];   // gt corner y
    __shared__ float s_gar[N_GT];       // gt area

    const int tid = threadIdx.x;

    // ---- CDNA5 async global->LDS staging of box_gt (3584 dwords) ----
    #pragma unroll
    for (int i = 0; i < (N_GT * 7) / TPB; ++i) {
        const int idx = i * TPB + tid;
        unsigned lds_addr = (unsigned)(size_t)(&s_raw[idx]);  // addrspace(3) offset in low 32 bits
        unsigned goff = (unsigned)(idx * 4);
        asm volatile("global_load_async_to_lds_b32 %0, %1, %2"
                     :: "v"(lds_addr), "v"(goff), "s"(box_gt)
                     : "memory");
    }
    asm volatile("s_wait_asynccnt 0" ::: "memory");
    __syncthreads();

    // ---- precompute GT corners + areas in LDS ----
    for (int j = tid; j < N_GT; j += TPB) {
        const float x   = s_raw[j*7+0], y  = s_raw[j*7+1];
        const float dx  = s_raw[j*7+3], dy = s_raw[j*7+4];
        const float yaw = s_raw[j*7+6];
        const float hx = 0.5f*dx, hy = 0.5f*dy;
        const float cs = cosf(yaw), sn = sinf(yaw);
        const float lx[4] = { hx, -hx, -hx,  hx };
        const float ly[4] = { hy,  hy, -hy, -hy };
        #pragma unroll
        for (int k = 0; k < 4; ++k) {
            s_gcx[j*4+k] = cs*lx[k] - sn*ly[k] + x;
            s_gcy[j*4+k] = sn*lx[k] + cs*ly[k] + y;
        }
        s_gar[j] = dx * dy;
    }
    __syncthreads();

    const int p = blockIdx.x * TPB + tid;

    // ---- pred corners in registers ----
    float pxr[4], pyr[4], areaA;
    {
        const float x   = box_pred[p*7+0], y  = box_pred[p*7+1];
        const float dx  = box_pred[p*7+3], dy = box_pred[p*7+4];
        const float yaw = box_pred[p*7+6];
        const float hx = 0.5f*dx, hy = 0.5f*dy;
        const float cs = cosf(yaw), sn = sinf(yaw);
        const float lx[4] = { hx, -hx, -hx,  hx };
        const float ly[4] = { hy,  hy, -hy, -hy };
        #pragma unroll
        for (int k = 0; k < 4; ++k) {
            pxr[k] = cs*lx[k] - sn*ly[k] + x;
            pyr[k] = sn*lx[k] + cs*ly[k] + y;
        }
        areaA = dx * dy;
    }

    float best = 0.0f;
    float sx[24], sy[24], sa[24];   // compact candidate points (scratch)

    for (int g = 0; g < N_GT; ++g) {
        float qx[4], qy[4];
        #pragma unroll
        for (int k = 0; k < 4; ++k) { qx[k] = s_gcx[g*4+k]; qy[k] = s_gcy[g*4+k]; }
        const float areaB = s_gar[g];

        int   m    = 0;
        float sumx = 0.f, sumy = 0.f;

        // 16 edge-edge intersections (same slot order as reference)
        #pragma unroll
        for (int i = 0; i < 4; ++i) {
            const float p1x = pxr[i], p1y = pyr[i];
            const float d1x = pxr[(i+1)&3] - p1x, d1y = pyr[(i+1)&3] - p1y;
            #pragma unroll
            for (int j = 0; j < 4; ++j) {
                const float q1x = qx[j], q1y = qy[j];
                const float d2x = qx[(j+1)&3] - q1x, d2y = qy[(j+1)&3] - q1y;
                const float dfx = q1x - p1x, dfy = q1y - p1y;
                const float den = d1x*d2y - d1y*d2x;
                const bool  ok  = fabsf(den) > EPS;
                const float sf  = ok ? den : 1.0f;
                const float t   = (dfx*d2y - dfy*d2x) / sf;
                const float u   = (dfx*d1y - dfy*d1x) / sf;
                if (ok && t >= 0.f && t <= 1.f && u >= 0.f && u <= 1.f) {
                    const float X = p1x + t*d1x, Y = p1y + t*d1y;
                    sx[m] = X; sy[m] = Y; sumx += X; sumy += Y; ++m;
                }
            }
        }
        // pred corners inside gt
        #pragma unroll
        for (int k = 0; k < 4; ++k) {
            bool in = true;
            #pragma unroll
            for (int j = 0; j < 4; ++j) {
                const float ax = qx[j], ay = qy[j];
                const float bxe = qx[(j+1)&3], bye = qy[(j+1)&3];
                const float cr = (bxe-ax)*(pyr[k]-ay) - (bye-ay)*(pxr[k]-ax);
                in = in && (cr >= -EPS);
            }
            if (in) { sx[m]=pxr[k]; sy[m]=pyr[k]; sumx+=pxr[k]; sumy+=pyr[k]; ++m; }
        }
        // gt corners inside pred
        #pragma unroll
        for (int k = 0; k < 4; ++k) {
            bool in = true;
            #pragma unroll
            for (int j = 0; j < 4; ++j) {
                const float ax = pxr[j], ay = pyr[j];
                const float bxe = pxr[(j+1)&3], bye = pyr[(j+1)&3];
                const float cr = (bxe-ax)*(qy[k]-ay) - (bye-ay)*(qx[k]-ax);
                in = in && (cr >= -EPS);
            }
            if (in) { sx[m]=qx[k]; sy[m]=qy[k]; sumx+=qx[k]; sumy+=qy[k]; ++m; }
        }

        float inter = 0.0f;
        if (m >= 3) {
            const float inv = 1.0f / (float)m;
            const float cx0 = sumx*inv, cy0 = sumy*inv;
            // stable insertion sort by polar angle (matches jnp.argsort semantics)
            for (int i = 0; i < m; ++i) {
                const float xx = sx[i], yy = sy[i];
                const float a  = atan2f(yy - cy0, xx - cx0);
                int j = i;
                while (j > 0 && sa[j-1] > a) {
                    sa[j] = sa[j-1]; sx[j] = sx[j-1]; sy[j] = sy[j-1]; --j;
                }
                sa[j] = a; sx[j] = xx; sy[j] = yy;
            }
            float acc = 0.f;
            for (int i = 0; i < m; ++i) {
                const int i2 = (i+1 == m) ? 0 : i+1;
                acc += (sx[i]-cx0)*(sy[i2]-cy0) - (sy[i]-cy0)*(sx[i2]-cx0);
            }
            inter = 0.5f * fabsf(acc);
        }
        const float uni = fmaxf(areaA + areaB - inter, EPS);
        best = fmaxf(best, inter / uni);
    }

    // ---- loss ----
    const float ip = iou_pred[p];
    float pr = 1.0f / (1.0f + expf(-ip));
    pr = fminf(fmaxf(pr, 1e-4f), 1.0f - 1e-4f);
    const float d  = pr - best;
    const float ad = fabsf(d);
    const float sl = (ad < 1.0f) ? 0.5f*d*d : ad - 0.5f;
    const float contrib = sl * (5.0f / (float)n_pred);

    // ---- exact-f32 wave-level sum via V_WMMA_F32_16X16X4_F32 (ones B-matrix) ----
    // A row m = [v_m, 0, v_{m+16}, 0]; B all ones -> D[m][n] = v_m + v_{m+16}.
    // Lanes 0-15 each hold rows 0..7 (sum of v0..7 + v16..23), lanes 16-31 rows 8..15.
    v2f A; A.x = contrib; A.y = 0.0f;
    v2f B; B.x = 1.0f;    B.y = 1.0f;
    v8f C = {0.f,0.f,0.f,0.f,0.f,0.f,0.f,0.f};
    v8f D = __builtin_amdgcn_wmma_f32_16x16x4_f32(
        false, A, false, B, (short)0, C, false, false);
    const float ws = D[0]+D[1]+D[2]+D[3]+D[4]+D[5]+D[6]+D[7];
    const unsigned lane = tid & 31u;
    if (lane == 0u || lane == 16u) {
        atomicAdd(out, ws);
    }
}

extern "C" void kernel_launch(void* const* d_in, const int* in_sizes, int n_in,
                              void* d_out, int out_size, void* d_ws, size_t ws_size,
                              hipStream_t stream) {
    const float* iou_pred = (const float*)d_in[0];
    const float* box_pred = (const float*)d_in[1];
    const float* box_gt   = (const float*)d_in[2];
    (void)n_in; (void)out_size; (void)d_ws; (void)ws_size;

    float* out = (float*)d_out;
    const int n_pred = in_sizes[0];   // 2048, multiple of 256 -> full waves (EXEC all-1s for WMMA)

    IouLoss_zero_kernel<<<1, 1, 0, stream>>>(out);
    IouLoss_main_kernel<<<n_pred / TPB, TPB, 0, stream>>>(iou_pred, box_pred, box_gt, out, n_pred);
}